// ModelNew_5909874999875
// MI455X (gfx1250) — compile-verified
//
#include <hip/hip_runtime.h>
#include <cstdint>
#include <cstddef>

// ---------------------------------------------------------------------------
// C[i,j] = A[i] * B[i,j]   (N=32768, M=4096, fp32)
// HBM-bound: 1 GiB moved -> ~46 us at 23.3 TB/s; FLOPs negligible.
// CDNA5 path: global_load_async_to_lds_b128 in GVS mode (SGPR base + 32-bit
// VGPR offset), 4-deep s_wait_asynccnt software pipeline per wave,
// ds_load_b128 consumption, non-temporal b128 stores.
// ---------------------------------------------------------------------------

typedef float v4f __attribute__((ext_vector_type(4)));

#define WAVES_PER_BLOCK 8
#define LANES           32
#define STAGES          4

// One 16B async DMA fill: global(base+off32) -> LDS. Tracked by ASYNCcnt.
// GVS addressing: mem_addr = SGPR64 + sext(VGPR32) (ISA 10.2). off32 < 2^29
// for this problem, so sign extension is benign.
__device__ __forceinline__ void async_fill_b128(uint64_t gbase, uint32_t goff,
                                                uint32_t lds_off) {
    asm volatile("global_load_async_to_lds_b128 %0, %1, %2"
                 :
                 : "v"(lds_off), "v"(goff), "s"(gbase)
                 : "memory");
}

__device__ __forceinline__ void wait_async_le3() {
    asm volatile("s_wait_asynccnt 0x3" ::: "memory");
}
__device__ __forceinline__ void wait_async_0() {
    asm volatile("s_wait_asynccnt 0x0" ::: "memory");
}
// WAR guard: LDS reads of a stage must complete before the async engine may
// overwrite it (async LDS writes are unordered vs DS-indexed ops).
__device__ __forceinline__ void wait_ds_0() {
    asm volatile("s_wait_dscnt 0x0" ::: "memory");
}

// Tuned kernel. Launcher guarantees: M/4 is a power of two, a 32-float4 tile
// never straddles a row, and total tiles divide evenly across all waves.
__global__ __launch_bounds__(256) void rowscale_async(const float* __restrict__ A,
                                                      const float* __restrict__ B,
                                                      v4f* __restrict__ C4,
                                                      int row_shift, // log2(M/4)
                                                      int nit)       // tiles/wave
{
    __shared__ v4f lbuf[WAVES_PER_BLOCK][STAGES][LANES];   // 16 KB / block

    const int tid  = threadIdx.x;
    const int wave = tid >> 5;
    const int lane = tid & 31;

    const int stride = (int)gridDim.x * WAVES_PER_BLOCK;     // total waves
    const int W      = (int)blockIdx.x * WAVES_PER_BLOCK + wave;

    const uint64_t gbase = (uint64_t)(uintptr_t)B;
    // Byte offset of this lane's 16B element in tile t: t*512 + lane*16.
    const uint32_t lane16      = (uint32_t)lane << 4;
    const uint32_t stride_b    = (uint32_t)stride << 9;      // bytes per tile step
    const uint32_t lds_base    = (uint32_t)(uintptr_t)&lbuf[wave][0][lane];
    const uint32_t lds_stage_b = (uint32_t)(LANES * sizeof(v4f));   // 512 B

    // ---- prologue: STAGES fills in flight --------------------------------
    {
        uint32_t goff = ((uint32_t)W << 9) + lane16;
#pragma unroll
        for (int s = 0; s < STAGES; ++s) {
            async_fill_b128(gbase, goff, lds_base + (uint32_t)s * lds_stage_b);
            goff += stride_b;
        }
    }

    // ---- steady state ----------------------------------------------------
    uint32_t goff_next = ((uint32_t)W << 9) + lane16 + (uint32_t)STAGES * stride_b;
    for (int i = 0; i < nit - STAGES; ++i) {
        const int t     = W + i * stride;
        const int stage = i & (STAGES - 1);

        wait_async_le3();                        // oldest fill landed
        v4f v = lbuf[wave][stage][lane];         // ds_load_b128

        // Wave-uniform row -> scalar s_load of A. Power-of-two M: shift only.
        const int ts  = __builtin_amdgcn_readfirstlane(t);
        const float a = A[(ts << 5) >> row_shift];

        __builtin_nontemporal_store(v * a, &C4[(size_t)t * LANES + lane]);

        // Stage reuse: ensure the ds_load above has drained, then refill.
        wait_ds_0();
        async_fill_b128(gbase, goff_next,
                        lds_base + (uint32_t)stage * lds_stage_b);
        goff_next += stride_b;
    }

    // ---- drain -----------------------------------------------------------
    wait_async_0();
#pragma unroll
    for (int e = 0; e < STAGES; ++e) {
        const int i     = (nit - STAGES) + e;
        const int t     = W + i * stride;
        const int stage = i & (STAGES - 1);

        v4f v = lbuf[wave][stage][lane];
        const int ts  = __builtin_amdgcn_readfirstlane(t);
        const float a = A[(ts << 5) >> row_shift];
        __builtin_nontemporal_store(v * a, &C4[(size_t)t * LANES + lane]);
    }
}

// Generic fallback for unexpected shapes.
__global__ void rowscale_generic(const float* __restrict__ A,
                                 const float* __restrict__ B,
                                 float* __restrict__ C,
                                 long long total, int m)
{
    long long i  = (long long)blockIdx.x * blockDim.x + threadIdx.x;
    long long gs = (long long)gridDim.x * blockDim.x;
    for (; i < total; i += gs) {
        C[i] = A[i / m] * B[i];
    }
}

extern "C" void kernel_launch(void* const* d_in, const int* in_sizes, int n_in,
                              void* d_out, int out_size, void* d_ws, size_t ws_size,
                              hipStream_t stream) {
    const float* A = (const float*)d_in[0];
    const float* B = (const float*)d_in[1];
    float*       C = (float*)d_out;

    const long long n     = in_sizes[0];             // rows
    const long long total = in_sizes[1];             // n * m
    const long long m     = (n > 0) ? total / n : 0; // cols

    const int       blocks     = 4096;               // 32768 waves
    const long long totalWaves = (long long)blocks * WAVES_PER_BLOCK;
    const long long tiles      = total / (4 * LANES);

    const bool tuned_ok =
        (m > 0) &&
        ((m & (m - 1)) == 0) &&                      // power-of-two M (shift rows)
        (m % (4 * LANES) == 0) &&                    // tile never straddles a row
        (total % (4 * LANES) == 0) &&
        (tiles % totalWaves == 0) &&
        (tiles / totalWaves >= STAGES + 1) &&
        (total * 4 < (1LL << 31));                   // 32-bit byte offsets valid

    if (tuned_ok) {
        int row_shift = 0;                           // log2(m/4)
        while ((1LL << row_shift) < (m / 4)) ++row_shift;
        const int nit = (int)(tiles / totalWaves);   // 32 for reference shape
        rowscale_async<<<blocks, 256, 0, stream>>>(A, B, (v4f*)C, row_shift, nit);
    } else {
        rowscale_generic<<<8192, 256, 0, stream>>>(A, B, C, total, (int)m);
    }
}